// PatientAttentionNet_35192962023826
// MI455X (gfx1250) — compile-verified
//
#include <hip/hip_runtime.h>
#include <hip/hip_bf16.h>
#include <math.h>

// ---------------------------------------------------------------------------
// GATConv (out_channels = 1), N=100000 nodes, E=6.4M edges, D=512.
//   x = F @ W                      (FP32 WMMA 16x16x4 matvec, LDS-staged)
//   logits = leaky(a_src[s]+a_dst[d]) over E+N (self-loops appended)
//   per-dst softmax via 3 passes: atomic max, atomic exp-sum, normalized scatter
// Workspace: x[N], m[N], denom[N] floats (1.2 MB).
// HBM traffic ~360 MB total (F 205 MB + 3x edge_index 154 MB); x/m/denom/out
// are L2-resident (1.6 MB << 192 MB), so scatter atomics stay on-chip.
// ---------------------------------------------------------------------------

#define GAT_D        512
#define TILE_ROWS    16
#define LDS_STRIDE   516   // 512 + 4 pad: stride % 64 == 4 -> conflict-free A reads
#define NEG_SLOPE    0.2f

typedef __attribute__((ext_vector_type(2))) float v2f;
typedef __attribute__((ext_vector_type(8))) float v8f;

__device__ __forceinline__ void atomicMaxFloat(float* addr, float val) {
    // valid because m[] is initialized to -inf; positive floats compare as ints,
    // negative floats compare reversed as uints.
    if (val >= 0.0f)
        atomicMax((int*)addr, __float_as_int(val));
    else
        atomicMin((unsigned int*)addr, __float_as_uint(val));
}

__device__ __forceinline__ float leaky(float z) {
    return z > 0.0f ? z : NEG_SLOPE * z;
}

// ---------------------------------------------------------------------------
// Kernel 0: init m = -inf, denom = 0, out = bias
// ---------------------------------------------------------------------------
__global__ void gat_init(float* __restrict__ m, float* __restrict__ denom,
                         float* __restrict__ out, const float* __restrict__ bias,
                         int n) {
    int i = blockIdx.x * blockDim.x + threadIdx.x;
    if (i < n) {
        m[i]     = __int_as_float(0xFF800000);  // -inf
        denom[i] = 0.0f;
        out[i]   = bias[0];
    }
}

// ---------------------------------------------------------------------------
// Kernel 1: x = F @ W via V_WMMA_F32_16X16X4_F32.
// Block = 128 threads (4 waves) handles 16 rows of F.
// LDS tile 16x512 (padded stride 516), W staged in LDS.
// Wave w covers K-range [w*128, w*128+128) in 32 WMMA steps of K=4;
// partial column-0 results reduced across waves through LDS.
// A frag (16x4 f32): lanes 0-15 hold {K,K+1} of row=lane, lanes 16-31 {K+2,K+3}.
// B frag (4x16 f32): col 0 only -> lane0={W[K],W[K+1]}, lane16={W[K+2],W[K+3]};
//   loaded in all lanes (LDS broadcast) and zeroed by cndmask, so no EXEC
//   save/restore in the hot loop and both ds_load_b64s issue back-to-back.
// D col 0: lane0 c[0..7]=rows0..7, lane16 c[0..7]=rows8..15.
// ---------------------------------------------------------------------------
__global__ __launch_bounds__(128) void gat_matvec(const float* __restrict__ F,
                                                  const float* __restrict__ W,
                                                  float* __restrict__ x, int n) {
    __shared__ float ldsA[TILE_ROWS * LDS_STRIDE];
    __shared__ float ldsW[GAT_D];
    __shared__ float partial[4][TILE_ROWS];
    __shared__ float pf[128];   // fallback reduction scratch (unused on WMMA path)

    const int tid = threadIdx.x;
    const int nb  = blockIdx.x * TILE_ROWS;

    // Stage W (2 KB)
    for (int i = tid; i < GAT_D; i += 128) ldsW[i] = W[i];

    // Stage F tile, coalesced float4 per thread per row
    const int c4 = tid * 4;
    for (int r = 0; r < TILE_ROWS; ++r) {
        int row = nb + r;
        float4 v = make_float4(0.f, 0.f, 0.f, 0.f);
        if (row < n) v = *reinterpret_cast<const float4*>(F + (size_t)row * GAT_D + c4);
        *reinterpret_cast<float4*>(&ldsA[r * LDS_STRIDE + c4]) = v;
    }
    __syncthreads();

#if __has_builtin(__builtin_amdgcn_wmma_f32_16x16x4_f32)
    const int  lane   = tid & 31;
    const int  wave   = tid >> 5;
    const int  rowA   = lane & 15;
    const int  koff   = (lane >> 4) * 2;
    const int  kbase  = wave * 128;
    const bool isCol0 = (rowA == 0);   // lanes 0 and 16 carry B column 0

    v8f c = {};
    for (int s = 0; s < 32; ++s) {
        const int k = kbase + s * 4;
        v2f a, b;
        a[0] = ldsA[rowA * LDS_STRIDE + k + koff];
        a[1] = ldsA[rowA * LDS_STRIDE + k + koff + 1];
        // unconditional LDS broadcast read + select (no EXEC divergence)
        const float b0 = ldsW[k + koff];
        const float b1 = ldsW[k + koff + 1];
        b[0] = isCol0 ? b0 : 0.0f;
        b[1] = isCol0 ? b1 : 0.0f;
        c = __builtin_amdgcn_wmma_f32_16x16x4_f32(false, a, false, b,
                                                  (short)0, c, false, false);
    }
    if (lane == 0) {
        #pragma unroll
        for (int j = 0; j < 8; ++j) partial[wave][j] = c[j];
    }
    if (lane == 16) {
        #pragma unroll
        for (int j = 0; j < 8; ++j) partial[wave][8 + j] = c[j];
    }
    __syncthreads();
    if (tid < TILE_ROWS) {
        int row = nb + tid;
        if (row < n)
            x[row] = partial[0][tid] + partial[1][tid] +
                     partial[2][tid] + partial[3][tid];
    }
#else
    // Scalar fallback: 16 rows x 8 K-slices of 64
    const int r  = tid >> 3;
    const int ks = (tid & 7) * 64;
    float acc = 0.0f;
    for (int k = 0; k < 64; ++k)
        acc += ldsA[r * LDS_STRIDE + ks + k] * ldsW[ks + k];
    pf[tid] = acc;
    __syncthreads();
    if (tid < TILE_ROWS) {
        float s = 0.0f;
        for (int j = 0; j < 8; ++j) s += pf[tid * 8 + j];
        int row = nb + tid;
        if (row < n) x[row] = s;
    }
#endif
}

// ---------------------------------------------------------------------------
// Edge passes. Work item gid in [0, E+N): gid < E -> edge, else self-loop.
// edge_index layout: ei[0..E) = src, ei[E..2E) = dst.
// ---------------------------------------------------------------------------
__device__ __forceinline__ void edge_pair(const int* __restrict__ ei, int gid,
                                          int E, int& s, int& d) {
    if (gid < E) { s = ei[gid]; d = ei[E + gid]; }
    else         { s = gid - E; d = s; }
}

__global__ void gat_pass_max(const int* __restrict__ ei, const float* __restrict__ x,
                             const float* __restrict__ attS, const float* __restrict__ attD,
                             float* __restrict__ m, int E, int n) {
    int gid = blockIdx.x * blockDim.x + threadIdx.x;
    if (gid >= E + n) return;
    int s, d; edge_pair(ei, gid, E, s, d);
    float z = leaky(x[s] * attS[0] + x[d] * attD[0]);
    atomicMaxFloat(&m[d], z);
}

__global__ void gat_pass_expsum(const int* __restrict__ ei, const float* __restrict__ x,
                                const float* __restrict__ attS, const float* __restrict__ attD,
                                const float* __restrict__ m, float* __restrict__ denom,
                                int E, int n) {
    int gid = blockIdx.x * blockDim.x + threadIdx.x;
    if (gid >= E + n) return;
    int s, d; edge_pair(ei, gid, E, s, d);
    float z = leaky(x[s] * attS[0] + x[d] * attD[0]);
    atomicAdd(&denom[d], __expf(z - m[d]));
}

__global__ void gat_pass_scatter(const int* __restrict__ ei, const float* __restrict__ x,
                                 const float* __restrict__ attS, const float* __restrict__ attD,
                                 const float* __restrict__ m, const float* __restrict__ denom,
                                 float* __restrict__ out, int E, int n) {
    int gid = blockIdx.x * blockDim.x + threadIdx.x;
    if (gid >= E + n) return;
    int s, d; edge_pair(ei, gid, E, s, d);
    float z = leaky(x[s] * attS[0] + x[d] * attD[0]);
    float alpha = __expf(z - m[d]) / denom[d];
    atomicAdd(&out[d], alpha * x[s]);
}

// ---------------------------------------------------------------------------
extern "C" void kernel_launch(void* const* d_in, const int* in_sizes, int n_in,
                              void* d_out, int out_size, void* d_ws, size_t ws_size,
                              hipStream_t stream) {
    const float* F    = (const float*)d_in[0];
    const int*   ei   = (const int*)  d_in[1];
    const float* W    = (const float*)d_in[2];
    const float* attS = (const float*)d_in[3];
    const float* attD = (const float*)d_in[4];
    const float* bias = (const float*)d_in[5];
    float*       out  = (float*)d_out;

    const int N = out_size;           // 100000
    const int E = in_sizes[1] / 2;    // 6400000

    float* x     = (float*)d_ws;      // N floats
    float* m     = x + N;             // N floats
    float* denom = m + N;             // N floats

    gat_init<<<(N + 255) / 256, 256, 0, stream>>>(m, denom, out, bias, N);
    gat_matvec<<<(N + TILE_ROWS - 1) / TILE_ROWS, 128, 0, stream>>>(F, W, x, N);

    const int tot = E + N;
    const int gb  = (tot + 255) / 256;
    gat_pass_max   <<<gb, 256, 0, stream>>>(ei, x, attS, attD, m, E, N);
    gat_pass_expsum<<<gb, 256, 0, stream>>>(ei, x, attS, attD, m, denom, E, N);
    gat_pass_scatter<<<gb, 256, 0, stream>>>(ei, x, attS, attD, m, denom, out, E, N);
}